// ClauseInferModule_28260884808446
// MI455X (gfx1250) — compile-verified
//
#include <hip/hip_runtime.h>
#include <stdint.h>

#define C_ 8
#define B_ 32
#define G_ 2048
#define S_ 32
#define L_ 3
#define GAMMA_ 0.01f
#define INV_GAMMA_ 100.0f
#define BG (B_*G_)
#define CBG (C_*B_*G_)
#define NCHUNK 16
#define G_PER_WG (G_/NCHUNK)    // 128 atoms per workgroup
#define THREADS 256
#define NWAVES (THREADS/32)

#define USE_ASYNC_STAGE 1       // gfx1250 global_load_async_to_lds path

__device__ __forceinline__ float wave_max32(float v) {
#pragma unroll
  for (int off = 16; off; off >>= 1)
    v = fmaxf(v, __shfl_xor(v, off, 32));
  return v;
}

// valid for non-negative floats: uint bit pattern is monotonic
__device__ __forceinline__ void atomic_max_pos_f32(unsigned* p, float v) {
  atomicMax(p, __float_as_uint(v));
}

__global__ void init_kernel(const float* __restrict__ x, float* __restrict__ R,
                            unsigned* __restrict__ maxes) {
  const int idx = blockIdx.x * blockDim.x + threadIdx.x;   // over C*B*G
  R[idx] = x[idx & (BG - 1)];                              // broadcast x over clause axis
  if (blockIdx.x == 0 && threadIdx.x < 64) maxes[threadIdx.x] = 0u;
}

// One workgroup = (clause c, chunk of 128 g's). Stage full R[c] (256KB) in LDS,
// transposed to [g][b] so the 32 batch-lanes gather from 32 distinct banks.
__global__ void __launch_bounds__(THREADS, 1)
clause_kernel(const float* __restrict__ R, const int* __restrict__ I,
              float* __restrict__ r_lse, unsigned* __restrict__ cmax) {
  __shared__ float Rl[G_ * B_];            // 256 KB of the 320 KB WGP LDS
  const int c     = blockIdx.x / NCHUNK;
  const int chunk = blockIdx.x - c * NCHUNK;
  const int t     = threadIdx.x;
  const int lane  = t & 31;                // batch index b
  const int wave  = t >> 5;
  const float* Rc = R + (size_t)c * BG;    // [B][G] row-major

  // ---- stage R[c] -> LDS transposed: Rl[g*32 + b] = Rc[b*G + g] ----
#if USE_ASYNC_STAGE
  for (int e = t; e < BG; e += THREADS) {  // 65536/256: no tail, EXEC all-1
    const int b = e >> 11;                 // G_ = 2048
    const int g = e & (G_ - 1);
    const float* gp = Rc + b * G_ + g;     // coalesced along g
    unsigned lds_off = (unsigned)(uintptr_t)&Rl[g * B_ + b];
    asm volatile("global_load_async_to_lds_b32 %0, %1, off"
                 :: "v"(lds_off), "v"(gp) : "memory");
  }
  asm volatile("s_wait_asynccnt 0" ::: "memory");
#else
  for (int e = t; e < BG; e += THREADS) {
    const int b = e >> 11;
    const int g = e & (G_ - 1);
    Rl[g * B_ + b] = Rc[b * G_ + g];
  }
#endif
  __syncthreads();

  const int g0 = chunk * G_PER_WG;
  float local_max = 0.0f;                  // lse values are always >= 0

  for (int gi = wave; gi < G_PER_WG; gi += NWAVES) {
    const int g = g0 + gi;
    const int* Ig = I + ((size_t)c * G_ + g) * (S_ * L_);  // uniform per wave

    float body[S_];
    float m = 0.0f;                        // body >= 0, so max ends up exact
#pragma unroll
    for (int s = 0; s < S_; ++s) {
      const int i0 = Ig[s * 3 + 0];
      const int i1 = Ig[s * 3 + 1];
      const int i2 = Ig[s * 3 + 2];
      const float v = Rl[i0 * B_ + lane] * Rl[i1 * B_ + lane] * Rl[i2 * B_ + lane];
      body[s] = v;
      m = fmaxf(m, v);
    }
    float sum = 0.0f;
#pragma unroll
    for (int s = 0; s < S_; ++s)
      sum += __expf((body[s] - m) * INV_GAMMA_);   // exponents in [-104, 0]
    const float lse = fmaf(GAMMA_, __logf(sum), m);

    r_lse[((size_t)c * B_ + lane) * G_ + g] = lse;
    local_max = fmaxf(local_max, lse);
  }

  local_max = wave_max32(local_max);
  if (lane == 0) atomic_max_pos_f32(&cmax[c], local_max);
}

// r' = (mc>1 ? r/mc : r); lse2 = pairwise gamma-logsumexp(R, r'); global max partial
__global__ void combine_kernel(const float* __restrict__ R, float* __restrict__ lse_buf,
                               const unsigned* __restrict__ cmax,
                               unsigned* __restrict__ gmax) {
  const int idx = blockIdx.x * blockDim.x + threadIdx.x;   // over C*B*G
  const int c   = idx >> 16;                               // / (B*G)
  const float mc = __uint_as_float(cmax[c]);
  float rv = lse_buf[idx];
  if (mc > 1.0f) rv /= mc;
  const float Rv = R[idx];
  const float hi = fmaxf(Rv, rv);
  const float lo = fminf(Rv, rv);
  const float l2 = fmaf(GAMMA_, __logf(1.0f + __expf((lo - hi) * INV_GAMMA_)), hi);
  lse_buf[idx] = l2;                                       // in-place
  const float wm = wave_max32(l2);
  if ((threadIdx.x & 31) == 0) atomic_max_pos_f32(gmax, wm);
}

__global__ void norm_kernel(const float* __restrict__ lse_buf,
                            const unsigned* __restrict__ gmax,
                            float* __restrict__ R) {
  const int idx = blockIdx.x * blockDim.x + threadIdx.x;
  const float m = __uint_as_float(*gmax);
  const float v = lse_buf[idx];
  R[idx] = (m > 1.0f) ? v / m : v;
}

extern "C" void kernel_launch(void* const* d_in, const int* in_sizes, int n_in,
                              void* d_out, int out_size, void* d_ws, size_t ws_size,
                              hipStream_t stream) {
  (void)in_sizes; (void)n_in; (void)out_size; (void)ws_size;
  const float* x = (const float*)d_in[0];   // [B,G] f32
  const int*   I = (const int*)d_in[1];     // [C,G,S,L] i32
  float* R = (float*)d_out;                 // [C,B,G] f32 — lives in d_out across steps
  float* lse_buf   = (float*)d_ws;                                   // C*B*G f32
  unsigned* maxes  = (unsigned*)((char*)d_ws + (size_t)CBG * sizeof(float)); // 64 slots

  init_kernel<<<CBG / THREADS, THREADS, 0, stream>>>(x, R, maxes);
  for (int step = 0; step < 4; ++step) {
    unsigned* cmax = maxes + step * (C_ + 1);
    unsigned* gmax = cmax + C_;
    clause_kernel <<<C_ * NCHUNK,  THREADS, 0, stream>>>(R, I, lse_buf, cmax);
    combine_kernel<<<CBG / THREADS, THREADS, 0, stream>>>(R, lse_buf, cmax, gmax);
    norm_kernel   <<<CBG / THREADS, THREADS, 0, stream>>>(lse_buf, gmax, R);
  }
}